// V1Column_36155034697745
// MI455X (gfx1250) — compile-verified
//
#include <hip/hip_runtime.h>
#include <stdint.h>

#define DTs   1.0f
#define MAXD  5
#define RB    5   // synaptic basis count

// ---------------------------------------------------------------- init
__global__ void k_init(float* ws, long long n) {
  long long i = (long long)blockIdx.x * blockDim.x + threadIdx.x;
  if (i < n) ws[i] = 0.f;
}

// ------------------------------------------------- pack spike floats -> bytes
__global__ void k_pack(const float* __restrict__ lgn_t, int n_lgn,
                       const float* __restrict__ bkg_t, int n_bkg,
                       unsigned char* __restrict__ spk, int pad_lgn, int total) {
  int i = blockIdx.x * blockDim.x + threadIdx.x;
  if (i >= total) return;
  unsigned char v = 0;
  if (i < pad_lgn) {
    if (i < n_lgn) v = (lgn_t[i] != 0.f) ? 1 : 0;
  } else {
    int j = i - pad_lgn;
    if (j < n_bkg) v = (bkg_t[j] != 0.f) ? 1 : 0;
  }
  spk[i] = v;
}

__device__ __forceinline__ void scatter5(float* __restrict__ ap, float w,
                                         const float* __restrict__ tp) {
#pragma unroll
  for (int j = 0; j < RB; ++j) {
    __hip_atomic_fetch_add(ap + j, w * tp[j], __ATOMIC_RELAXED,
                           __HIP_MEMORY_SCOPE_AGENT);
  }
}

// ---------------------------------------------------------------- scatter
// Fused edge scatter for rec/in/bkg populations. LGN+BKG spike bytes are
// staged into LDS with CDNA5 async loads (GLOBAL_LOAD_ASYNC_TO_LDS_B128).
// Recurrent spikes are gathered from a u8 delay ring (244 KB working set).
__global__ void k_scatter(
    const unsigned char* __restrict__ zbytes, // [MAXD][N] u8 delay ring
    float* __restrict__ acc,                  // [N][RB] accumulator
    const unsigned char* __restrict__ spk,    // packed spike bytes (global)
    int staged_bytes, int pad_lgn,
    const float* __restrict__ w_rec, const float* __restrict__ tau_rec,
    const int* __restrict__ rec_post, const int* __restrict__ rec_col,
    const float* __restrict__ w_in, const float* __restrict__ tau_in,
    const int* __restrict__ in_post, const int* __restrict__ in_pre,
    const float* __restrict__ w_bkg, const float* __restrict__ tau_bkg,
    const int* __restrict__ bkg_post, const int* __restrict__ bkg_pre,
    int N, int E_rec, int E_in, int E_bkg, int tbase /* == (t+9)%MAXD */) {
  extern __shared__ unsigned char smem[];

  // ---- CDNA5 async stage: global -> LDS, tracked by ASYNCcnt ----
  unsigned lds_base =
      (unsigned)(uintptr_t)(__attribute__((address_space(3))) unsigned char*)smem;
  unsigned long long gbase = (unsigned long long)(uintptr_t)spk;
  int nvec = staged_bytes >> 4;  // 16B per async b128 transfer per lane
  for (int i = threadIdx.x; i < nvec; i += blockDim.x) {
    unsigned loff = lds_base + ((unsigned)i << 4);
    unsigned long long ga = gbase + ((unsigned long long)i << 4);
    asm volatile("global_load_async_to_lds_b128 %0, %1, off"
                 :: "v"(loff), "v"(ga) : "memory");
  }
  asm volatile("s_wait_asynccnt 0" ::: "memory");
  __syncthreads();

  int N2 = 2 * N, N3 = 3 * N, N4 = 4 * N;   // wave-uniform SALU
  int E_total = E_rec + E_in + E_bkg;
  int stride = gridDim.x * blockDim.x;
  for (int e = blockIdx.x * blockDim.x + threadIdx.x; e < E_total; e += stride) {
    if (e < E_rec) {
      int c = rec_col[e];
      // delay d in [0,4]: branch-free, no integer divide
      int d = (c >= N) + (c >= N2) + (c >= N3) + (c >= N4);
      int src = c - d * N;
      int slot = tbase - d; slot += (slot < 0) ? MAXD : 0;
      if (zbytes[(long long)slot * N + src]) {   // spike is exactly 0/1
        scatter5(acc + (long long)rec_post[e] * RB, w_rec[e],
                 tau_rec + (long long)e * RB);
      }
    } else if (e < E_rec + E_in) {
      int i = e - E_rec;
      if (smem[in_pre[i]]) {         // LDS gather of staged LGN spikes
        scatter5(acc + (long long)in_post[i] * RB, w_in[i],
                 tau_in + (long long)i * RB);
      }
    } else {
      int i = e - E_rec - E_in;
      if (smem[pad_lgn + bkg_pre[i]]) {
        scatter5(acc + (long long)bkg_post[i] * RB, w_bkg[i],
                 tau_bkg + (long long)i * RB);
      }
    }
  }
}

// ---------------------------------------------------------------- update
__global__ void k_update(
    unsigned char* __restrict__ zbytes, float* __restrict__ v,
    float* __restrict__ r, float* __restrict__ asc,
    float* __restrict__ psc_rise, float* __restrict__ psc,
    float* __restrict__ acc,
    const float* __restrict__ v_th, const float* __restrict__ e_l,
    const float* __restrict__ v_reset, const float* __restrict__ g,
    const float* __restrict__ decay, const float* __restrict__ current_factor,
    const float* __restrict__ t_ref, const float* __restrict__ asc_amps,
    const float* __restrict__ kk, const float* __restrict__ syn_decay,
    const float* __restrict__ psc_initial,
    float* __restrict__ out_t, int N, int slot_prev, int slot_new) {
  int n = blockIdx.x * blockDim.x + threadIdx.x;
  if (n >= N) return;

  float pz = (float)zbytes[(long long)slot_prev * N + n];
  long long b = (long long)n * RB;
  long long a2 = 2ll * n;

  float sd[RB], pi[RB];
#pragma unroll
  for (int j = 0; j < RB; ++j) { sd[j] = syn_decay[j]; pi[j] = psc_initial[j]; }

  float ic = 0.f, np[RB], npr[RB];
#pragma unroll
  for (int j = 0; j < RB; ++j) {
    float pj  = psc[b + j];
    float prj = psc_rise[b + j];
    float aj  = acc[b + j];
    ic += pj;                                   // input current uses OLD psc
    npr[j] = prj * sd[j] + aj * pi[j];
    np[j]  = pj * sd[j] + DTs * sd[j] * prj;    // uses OLD psc_rise
    acc[b + j] = 0.f;                           // clear for next step
  }

  float a0 = asc[a2], a1 = asc[a2 + 1];
  float asc_c = a0 + a1;                        // OLD asc
  float new_r = fmaxf(0.f, r[n] + pz * t_ref[n] - DTs);
  float na0 = __expf(-DTs * kk[a2])     * a0 + pz * asc_amps[a2];
  float na1 = __expf(-DTs * kk[a2 + 1]) * a1 + pz * asc_amps[a2 + 1];

  float el = e_l[n], vth = v_th[n];
  float c1 = ic + asc_c + g[n] * el;
  float nv = decay[n] * v[n] + current_factor[n] * c1;
  if (pz > 0.5f) nv = v_reset[n];
  float vsc = (nv - vth) / (vth - el);
  float nz = (vsc > 0.f) ? 1.f : 0.f;
  if (new_r > 0.f) nz = 0.f;

  v[n] = nv; r[n] = new_r;
  asc[a2] = na0; asc[a2 + 1] = na1;
#pragma unroll
  for (int j = 0; j < RB; ++j) { psc_rise[b + j] = npr[j]; psc[b + j] = np[j]; }
  zbytes[(long long)slot_new * N + n] = (unsigned char)nz;
  out_t[n] = nz;
}

// ---------------------------------------------------------------- launcher
extern "C" void kernel_launch(void* const* d_in, const int* in_sizes, int n_in,
                              void* d_out, int out_size, void* d_ws,
                              size_t ws_size, hipStream_t stream) {
  (void)n_in; (void)ws_size;
  const float* lgn        = (const float*)d_in[0];
  const float* bkg        = (const float*)d_in[1];
  const float* w_rec      = (const float*)d_in[2];
  const float* tau_rec    = (const float*)d_in[3];
  const int*   rec_post   = (const int*)d_in[4];
  const int*   rec_col    = (const int*)d_in[5];
  const float* w_in       = (const float*)d_in[6];
  const float* tau_in     = (const float*)d_in[7];
  const int*   in_post    = (const int*)d_in[8];
  const int*   in_pre     = (const int*)d_in[9];
  const float* w_bkg      = (const float*)d_in[10];
  const float* tau_bkg    = (const float*)d_in[11];
  const int*   bkg_post   = (const int*)d_in[12];
  const int*   bkg_pre    = (const int*)d_in[13];
  const float* v_th       = (const float*)d_in[14];
  const float* e_l        = (const float*)d_in[15];
  const float* v_reset    = (const float*)d_in[16];
  const float* g          = (const float*)d_in[17];
  const float* decay      = (const float*)d_in[18];
  const float* cur_factor = (const float*)d_in[19];
  const float* t_ref      = (const float*)d_in[20];
  const float* asc_amps   = (const float*)d_in[21];
  const float* kk         = (const float*)d_in[22];
  const float* syn_decay  = (const float*)d_in[23];
  const float* psc_init   = (const float*)d_in[24];
  float* out = (float*)d_out;

  int N     = in_sizes[14];
  int T     = out_size / N;            // B==1 in reference setup
  int N_LGN = in_sizes[0] / T;
  int N_BKG = in_sizes[1] / T;
  int E_rec = in_sizes[2];
  int E_in  = in_sizes[6];
  int E_bkg = in_sizes[10];

  // workspace layout: 19N floats of state, then u8 delay ring, then spike bytes
  float* f = (float*)d_ws;
  float* v        = f; f += N;
  float* r        = f; f += N;
  float* asc      = f; f += 2 * (size_t)N;
  float* psc_rise = f; f += (size_t)RB * N;
  float* psc      = f; f += (size_t)RB * N;
  float* acc      = f; f += (size_t)RB * N;
  unsigned char* zbytes = (unsigned char*)f;
  size_t zpad = ((size_t)MAXD * N + 15) & ~(size_t)15;
  unsigned char* spk = zbytes + zpad;

  int pad_lgn = (N_LGN + 15) & ~15;
  int pad_bkg = (N_BKG + 15) & ~15;
  int staged  = pad_lgn + pad_bkg;     // ~17.6 KB -> dynamic LDS

  // zero all state floats + the byte ring (zpad is 16B-aligned)
  long long ztot = (long long)N * (1 + 1 + 2 + 3 * RB) + (long long)(zpad / 4);
  k_init<<<(unsigned)((ztot + 255) / 256), 256, 0, stream>>>((float*)d_ws, ztot);

  int pblk = (staged + 255) / 256;
  int ublk = (N + 255) / 256;
  int sblk = 1536;                     // grid-stride over 2.3M edges

  for (int t = 0; t < T; ++t) {
    k_pack<<<pblk, 256, 0, stream>>>(lgn + (size_t)t * N_LGN, N_LGN,
                                     bkg + (size_t)t * N_BKG, N_BKG,
                                     spk, pad_lgn, staged);
    int tbase = (t + 9) % MAXD;
    k_scatter<<<sblk, 256, staged, stream>>>(
        zbytes, acc, spk, staged, pad_lgn,
        w_rec, tau_rec, rec_post, rec_col,
        w_in, tau_in, in_post, in_pre,
        w_bkg, tau_bkg, bkg_post, bkg_pre,
        N, E_rec, E_in, E_bkg, tbase);
    k_update<<<ublk, 256, 0, stream>>>(
        zbytes, v, r, asc, psc_rise, psc, acc,
        v_th, e_l, v_reset, g, decay, cur_factor, t_ref, asc_amps, kk,
        syn_decay, psc_init, out + (size_t)t * N, N,
        (t + 4) % MAXD, t % MAXD);
  }
}